// GATSkillEncoder_53927609368622
// MI455X (gfx1250) — compile-verified
//
#include <hip/hip_runtime.h>
#include <math.h>

#define EMB 128
#define NEG_SLOPE 0.2f

typedef float v2f __attribute__((ext_vector_type(2)));
typedef float v8f __attribute__((ext_vector_type(8)));

__device__ __forceinline__ void atomicMaxFloat(float* addr, float v) {
    // Works for all finite floats + -inf init: positive path via signed int max,
    // negative path via unsigned int min.
    if (v >= 0.0f) atomicMax((int*)addr, __float_as_int(v));
    else           atomicMin((unsigned int*)addr, __float_as_uint(v));
}

// ---------------- init: zero output/denoms, -inf maxes ----------------
__global__ void gat_init_kernel(float* __restrict__ o, float* __restrict__ deg,
                                float* __restrict__ lw, float* __restrict__ amax,
                                float* __restrict__ denom, int N) {
    int i = blockIdx.x * blockDim.x + threadIdx.x;
    if (i < N * EMB) o[i] = 0.0f;
    if (i < N) { deg[i] = 0.0f; lw[i] = 0.0f; }
    if (i < 2 * N) { amax[i] = -INFINITY; denom[i] = 0.0f; }
}

// ---------------- degree + self-loop weight sums ----------------
__global__ void gat_deg_kernel(const int* __restrict__ dst, const float* __restrict__ ew,
                               float* __restrict__ deg, float* __restrict__ lw, int E) {
    int e = blockIdx.x * blockDim.x + threadIdx.x;
    if (e >= E) return;
    int d = dst[e];
    atomicAdd(&deg[d], 1.0f);
    atomicAdd(&lw[d], ew[e]);
}

// ---------------- xl = h@Wl, xr = h@Wr via V_WMMA_F32_16X16X4_F32 ----------------
// One wave32 computes one 16x16 output tile; 32 WMMA steps over K=128.
__global__ __launch_bounds__(256)
void gat_gemm_kernel(const float* __restrict__ h, const float* __restrict__ Wl,
                     const float* __restrict__ Wr, float* __restrict__ xl,
                     float* __restrict__ xr, int N) {
    int lane = threadIdx.x & 31;
    int wave = (blockIdx.x * blockDim.x + threadIdx.x) >> 5;
    int rowTiles = (N + 15) >> 4;
    if (wave >= rowTiles * 16) return;     // uniform per-wave exit: EXEC stays all-ones
    int rowTile = wave >> 4;
    int t = wave & 15;                      // 8 col-tiles for Wl, 8 for Wr
    const float* W = (t & 8) ? Wr : Wl;
    float*      out = (t & 8) ? xr : xl;
    int ncol = (t & 7) * 16;
    int half = lane >> 4;                   // lanes 16-31 carry K+2/K+3 (A) rows M+8 (C/D)
    int m    = lane & 15;
    int arow = rowTile * 16 + m;
    if (arow > N - 1) arow = N - 1;         // clamp (N%16==0 here, harmless)
    int col  = ncol + m;
    int koff = half * 2;

    v8f c = {0.f, 0.f, 0.f, 0.f, 0.f, 0.f, 0.f, 0.f};
#pragma unroll
    for (int k = 0; k < EMB; k += 4) {
        // A 16x4: lane holds {h[arow][k+koff], h[arow][k+koff+1]}
        v2f a = *(const v2f*)(h + (size_t)arow * EMB + k + koff);
        // B 4x16: lane holds {W[k+koff][col], W[k+koff+1][col]}
        const float* wp = W + (size_t)(k + koff) * EMB + col;
        v2f b; b.x = wp[0]; b.y = wp[EMB];
        c = __builtin_amdgcn_wmma_f32_16x16x4_f32(false, a, false, b, (short)0, c,
                                                  false, false);
    }
    // C/D layout: VGPR j -> row rowTile*16 + j + half*8, col = ncol + (lane&15)
#pragma unroll
    for (int j = 0; j < 8; ++j) {
        int r = rowTile * 16 + j + half * 8;
        if (r < N) out[(size_t)r * EMB + col] = c[j];
    }
}

// ---------------- per-edge attention logits + scatter max ----------------
// One wave per edge; lane covers 4 features (lanes 0-15 -> head0, 16-31 -> head1).
__global__ void gat_alpha_kernel(const float* __restrict__ xl, const float* __restrict__ xr,
                                 const int* __restrict__ src, const int* __restrict__ dst,
                                 const float* __restrict__ ew, const float* __restrict__ deg,
                                 const float* __restrict__ lw, const float* __restrict__ We,
                                 const float* __restrict__ att, float* __restrict__ alpha,
                                 float* __restrict__ amax, int E, int N) {
    int wave = (blockIdx.x * blockDim.x + threadIdx.x) >> 5;
    int lane = threadIdx.x & 31;
    if (wave >= E + N) return;
    int s, d; float w;
    if (wave < E) { s = src[wave]; d = dst[wave]; w = ew[wave]; }
    else { int i = wave - E; s = i; d = i; w = lw[i] / fmaxf(deg[i], 1.0f); }

    const float4 xs = *(const float4*)(xl + (size_t)s * EMB + lane * 4);
    const float4 xd = *(const float4*)(xr + (size_t)d * EMB + lane * 4);
    const float4 we = *(const float4*)(We + lane * 4);
    const float4 at = *(const float4*)(att + lane * 4);

    float p = 0.0f, v;
    v = xs.x + xd.x + w * we.x; v = (v >= 0.f) ? v : NEG_SLOPE * v; p += v * at.x;
    v = xs.y + xd.y + w * we.y; v = (v >= 0.f) ? v : NEG_SLOPE * v; p += v * at.y;
    v = xs.z + xd.z + w * we.z; v = (v >= 0.f) ? v : NEG_SLOPE * v; p += v * at.z;
    v = xs.w + xd.w + w * we.w; v = (v >= 0.f) ? v : NEG_SLOPE * v; p += v * at.w;

    // reduce within each half-wave (16 lanes = one head)
    for (int msk = 8; msk >= 1; msk >>= 1) p += __shfl_xor(p, msk, 32);

    if ((lane & 15) == 0) {
        int head = lane >> 4;
        alpha[(size_t)wave * 2 + head] = p;
        atomicMaxFloat(&amax[(size_t)d * 2 + head], p);
    }
}

// ---------------- exp + scatter denominator ----------------
__global__ void gat_expsum_kernel(const int* __restrict__ dst, float* __restrict__ alpha,
                                  const float* __restrict__ amax, float* __restrict__ denom,
                                  int E, int N) {
    int idx = blockIdx.x * blockDim.x + threadIdx.x;
    if (idx >= 2 * (E + N)) return;
    int e = idx >> 1, head = idx & 1;
    int d = (e < E) ? dst[e] : (e - E);
    float ex = __expf(alpha[idx] - amax[(size_t)d * 2 + head]);
    alpha[idx] = ex;   // overwrite logits with exp for the aggregate pass
    atomicAdd(&denom[(size_t)d * 2 + head], ex);
}

// ---------------- weighted scatter-add of messages ----------------
// One wave per edge; lane handles 4 features with float atomics into L2.
__global__ void gat_aggregate_kernel(const float* __restrict__ xl, const int* __restrict__ src,
                                     const int* __restrict__ dst, const float* __restrict__ alpha,
                                     const float* __restrict__ denom, float* __restrict__ o,
                                     int E, int N) {
    int wave = (blockIdx.x * blockDim.x + threadIdx.x) >> 5;
    int lane = threadIdx.x & 31;
    if (wave >= E + N) return;
    int s, d;
    if (wave < E) { s = src[wave]; d = dst[wave]; }
    else { s = d = wave - E; }
    int head = lane >> 4;
    float a = alpha[(size_t)wave * 2 + head] / denom[(size_t)d * 2 + head];
    const float4 m = *(const float4*)(xl + (size_t)s * EMB + lane * 4);
    float* op = o + (size_t)d * EMB + lane * 4;
    atomicAdd(op + 0, m.x * a);
    atomicAdd(op + 1, m.y * a);
    atomicAdd(op + 2, m.z * a);
    atomicAdd(op + 3, m.w * a);
}

// ---------------- bias + ELU ----------------
__global__ void gat_finish_kernel(float* __restrict__ o, const float* __restrict__ bias, int N) {
    int i = blockIdx.x * blockDim.x + threadIdx.x;
    if (i >= N * EMB) return;
    float x = o[i] + bias[i & (EMB - 1)];
    o[i] = (x > 0.0f) ? x : (__expf(x) - 1.0f);
}

extern "C" void kernel_launch(void* const* d_in, const int* in_sizes, int n_in,
                              void* d_out, int out_size, void* d_ws, size_t ws_size,
                              hipStream_t stream) {
    const int*   edge_index = (const int*)d_in[0];   // [2, E] int32
    const float* ew   = (const float*)d_in[1];
    const float* emb  = (const float*)d_in[2];
    const float* Wl   = (const float*)d_in[3];       // [2,128,128]
    const float* Wr   = (const float*)d_in[4];
    const float* We   = (const float*)d_in[5];       // [2,1,128]
    const float* att  = (const float*)d_in[6];       // [2,2,64] = [2,128]
    const float* bias = (const float*)d_in[7];       // [2,128]

    const int E = in_sizes[0] / 2;
    const int N = in_sizes[2] / EMB;
    const int* src = edge_index;
    const int* dst = edge_index + E;

    // workspace carve-up (floats)
    float* ws    = (float*)d_ws;
    float* xl    = ws;
    float* xr    = xl + (size_t)N * EMB;
    float* hbuf  = xr + (size_t)N * EMB;
    float* alpha = hbuf + (size_t)N * EMB;
    float* amax  = alpha + (size_t)2 * (E + N);
    float* denom = amax + (size_t)2 * N;
    float* deg   = denom + (size_t)2 * N;
    float* lw    = deg + N;

    const int BT = 256;
    const int rowTiles = (N + 15) / 16;
    const int gemmBlocks = (rowTiles * 16 + 7) / 8;          // 8 waves / block
    const int edgeWaveBlocks = ((E + N) * 32 + BT - 1) / BT; // 1 wave / edge
    const int nodeBlocks = (N * EMB + BT - 1) / BT;

    for (int l = 0; l < 2; ++l) {
        const float* hin = (l == 0) ? emb : hbuf;
        float* o = (l == 0) ? hbuf : (float*)d_out;
        const float* Wl_l = Wl + (size_t)l * EMB * EMB;
        const float* Wr_l = Wr + (size_t)l * EMB * EMB;
        const float* We_l = We + (size_t)l * EMB;
        const float* at_l = att + (size_t)l * EMB;
        const float* b_l  = bias + (size_t)l * EMB;

        gat_init_kernel<<<nodeBlocks, BT, 0, stream>>>(o, deg, lw, amax, denom, N);
        gat_deg_kernel<<<(E + BT - 1) / BT, BT, 0, stream>>>(dst, ew, deg, lw, E);
        gat_gemm_kernel<<<gemmBlocks, BT, 0, stream>>>(hin, Wl_l, Wr_l, xl, xr, N);
        gat_alpha_kernel<<<edgeWaveBlocks, BT, 0, stream>>>(xl, xr, src, dst, ew, deg, lw,
                                                            We_l, at_l, alpha, amax, E, N);
        gat_expsum_kernel<<<(2 * (E + N) + BT - 1) / BT, BT, 0, stream>>>(dst, alpha, amax,
                                                                          denom, E, N);
        gat_aggregate_kernel<<<edgeWaveBlocks, BT, 0, stream>>>(xl, src, dst, alpha, denom,
                                                                o, E, N);
        gat_finish_kernel<<<nodeBlocks, BT, 0, stream>>>(o, b_l, N);
    }
}